// ConditionAwareAdaIN_68925635166853
// MI455X (gfx1250) — compile-verified
//
#include <hip/hip_runtime.h>

typedef __attribute__((ext_vector_type(2))) float v2f;
typedef __attribute__((ext_vector_type(4))) float v4f;
typedef __attribute__((ext_vector_type(8))) float v8f;

constexpr int NB    = 16;    // batch
constexpr int NC    = 256;   // channels
constexpr int NL    = 1024;  // length
constexpr int DIMU  = 64;
constexpr int QEMB  = 64;
constexpr int TWOC  = 512;
constexpr int INTER = 4096;  // DIMU * QEMB
constexpr float EPS = 1e-5f;

constexpr int LT = 128;   // L-tile per block in the fused kernel
constexpr int MS = 68;    // LDS row stride for M tiles (pad: banks step 4/lane)
constexpr int ES = 144;   // LDS row stride for E tile (2*ES % 64 == 32 -> halves disjoint)

// ---------------------------------------------------------------------------
// Stage 1: M[b,c,q] = sum_u u_i[b,u] * W[c, u*64 + q]   (B=16 accumulators/thread)
// Thread layout: one thread per (c,q); q fastest -> W reads fully coalesced.
// ---------------------------------------------------------------------------
__global__ __launch_bounds__(256) void caain_style_u(const float* __restrict__ u_i,
                                                     const float* __restrict__ W,
                                                     float* __restrict__ M) {
  __shared__ float su[NB * DIMU];
  const int t = threadIdx.x;
  for (int i = t; i < NB * DIMU; i += 256) su[i] = u_i[i];
  __syncthreads();

  const int idx = blockIdx.x * 256 + t;   // 0 .. TWOC*QEMB-1
  const int c = idx >> 6;
  const int q = idx & 63;

  float acc[NB];
#pragma unroll
  for (int b = 0; b < NB; ++b) acc[b] = 0.f;

  const float* wrow = W + c * INTER + q;
#pragma unroll 4
  for (int u = 0; u < DIMU; ++u) {
    const float w = wrow[u * QEMB];
#pragma unroll
    for (int b = 0; b < NB; ++b) acc[b] = fmaf(su[b * DIMU + u], w, acc[b]);
  }
#pragma unroll
  for (int b = 0; b < NB; ++b) M[b * (TWOC * QEMB) + idx] = acc[b];
}

// ---------------------------------------------------------------------------
// Stage 2: per-(b,c) mean / rstd over L (one 256-thread block per row)
// ---------------------------------------------------------------------------
__global__ __launch_bounds__(256) void caain_stats(const float* __restrict__ x,
                                                   float* __restrict__ mean,
                                                   float* __restrict__ rstd) {
  const int row = blockIdx.x;                 // b*NC + c
  const float* xr = x + (size_t)row * NL;
  const int t = threadIdx.x;

  v4f v = *(const v4f*)(xr + t * 4);
  float s  = v.x + v.y + v.z + v.w;
  float ss = v.x * v.x + v.y * v.y + v.z * v.z + v.w * v.w;

  __shared__ float sh_s[256];
  __shared__ float sh_q[256];
  sh_s[t] = s;
  sh_q[t] = ss;
  __syncthreads();
#pragma unroll
  for (int off = 128; off > 0; off >>= 1) {
    if (t < off) {
      sh_s[t] += sh_s[t + off];
      sh_q[t] += sh_q[t + off];
    }
    __syncthreads();
  }
  if (t == 0) {
    const float m   = sh_s[0] * (1.f / NL);
    const float var = sh_q[0] * (1.f / NL) - m * m;
    mean[row] = m;
    rstd[row] = rsqrtf(var + EPS);
  }
}

// ---------------------------------------------------------------------------
// Stage 3 (fused): per block -> channel tile [c0,c0+16) x L tile [l0,l0+128).
// Each wave owns one 16-column slice; computes BOTH the gamma tile (rows c0..)
// and beta tile (rows c0+256..) as 16 chained V_WMMA_F32_16X16X4_F32 each,
// then applies InstanceNorm + (1+gamma)*nx + beta in registers.
// ---------------------------------------------------------------------------
__global__ __launch_bounds__(256) void caain_main(const float* __restrict__ x,
                                                  const float* __restrict__ e_qid,
                                                  const float* __restrict__ t_in,
                                                  const float* __restrict__ V,
                                                  const float* __restrict__ bias,
                                                  const float* __restrict__ M,
                                                  const float* __restrict__ mean,
                                                  const float* __restrict__ rstd,
                                                  float* __restrict__ out) {
  __shared__ float sMg[16 * MS];
  __shared__ float sMb[16 * MS];
  __shared__ float sE[QEMB * ES];

  const int b   = blockIdx.z;
  const int c0  = blockIdx.y * 16;
  const int l0  = blockIdx.x * LT;
  const int tid = threadIdx.x;

  // Stage M tiles (each is a contiguous 16*64 float region of M).
  const float* Mg_src = M + b * (TWOC * QEMB) + c0 * QEMB;
  const float* Mb_src = Mg_src + NC * QEMB;     // +256 rows (beta half)
  for (int i = tid; i < 16 * QEMB; i += 256) {
    const int r = i >> 6, q = i & 63;
    sMg[r * MS + q] = Mg_src[i];
    sMb[r * MS + q] = Mb_src[i];
  }
  // Stage E tile: rows q=0..63, cols [l0, l0+128), float4 coalesced.
  const float* E_src = e_qid + (size_t)b * (QEMB * NL) + l0;
  for (int i = tid * 4; i < QEMB * LT; i += 256 * 4) {
    const int q = i >> 7;           // LT == 128
    const int j = i & (LT - 1);
    v4f ev = *(const v4f*)(E_src + q * NL + j);
    *(v4f*)(sE + q * ES + j) = ev;
  }
  __syncthreads();

  const int wave     = tid >> 5;
  const int lane     = tid & 31;
  const int halflane = lane & 15;
  const int hi       = lane >> 4;       // 0 | 1
  const int koff     = hi * 2;          // A/B: lanes 16-31 carry K+2
  const int col      = wave * 16 + halflane;

  v8f accg = {0.f, 0.f, 0.f, 0.f, 0.f, 0.f, 0.f, 0.f};
  v8f accb = {0.f, 0.f, 0.f, 0.f, 0.f, 0.f, 0.f, 0.f};

#pragma unroll
  for (int k = 0; k < 16; ++k) {
    const int q0 = 4 * k + koff;
    v2f ag, ab, bb;
    ag.x = sMg[halflane * MS + q0];
    ag.y = sMg[halflane * MS + q0 + 1];
    ab.x = sMb[halflane * MS + q0];
    ab.y = sMb[halflane * MS + q0 + 1];
    bb.x = sE[q0 * ES + col];
    bb.y = sE[(q0 + 1) * ES + col];
    accg = __builtin_amdgcn_wmma_f32_16x16x4_f32(false, ag, false, bb, (short)0,
                                                 accg, false, false);
    accb = __builtin_amdgcn_wmma_f32_16x16x4_f32(false, ab, false, bb, (short)0,
                                                 accb, false, false);
  }

  // Epilogue. C/D layout: VGPR v, lanes0-15 -> row v, lanes16-31 -> row v+8.
  const int l = l0 + col;
  const float tval = t_in[b * NL + l];
#pragma unroll
  for (int v = 0; v < 8; ++v) {
    const int cg = c0 + v + hi * 8;                    // gamma channel 0..255
    const float gamma = accg[v] + tval * V[cg] + bias[cg];
    const float beta  = accb[v] + tval * V[cg + NC] + bias[cg + NC];
    const int   mrow  = b * NC + cg;
    const size_t xi   = (size_t)mrow * NL + l;
    const float nx = (x[xi] - mean[mrow]) * rstd[mrow];
    out[xi] = (1.f + gamma) * nx + beta;
  }
}

// ---------------------------------------------------------------------------
extern "C" void kernel_launch(void* const* d_in, const int* in_sizes, int n_in,
                              void* d_out, int out_size, void* d_ws, size_t ws_size,
                              hipStream_t stream) {
  const float* x     = (const float*)d_in[0];
  const float* u_i   = (const float*)d_in[1];
  const float* e_qid = (const float*)d_in[2];
  const float* t     = (const float*)d_in[3];
  const float* W     = (const float*)d_in[4];
  const float* V     = (const float*)d_in[5];
  const float* bias  = (const float*)d_in[6];
  float* out = (float*)d_out;

  float* M    = (float*)d_ws;                       // 16*512*64 floats = 2 MB
  float* mean = M + NB * TWOC * QEMB;               // 4096 floats
  float* rstd = mean + NB * NC;                     // 4096 floats

  caain_style_u<<<(TWOC * QEMB) / 256, 256, 0, stream>>>(u_i, W, M);
  caain_stats<<<NB * NC, 256, 0, stream>>>(x, mean, rstd);
  caain_main<<<dim3(NL / LT, NC / 16, NB), 256, 0, stream>>>(
      x, e_qid, t, V, bias, M, mean, rstd, out);
}